// NADE_31525059953050
// MI455X (gfx1250) — compile-verified
//
#include <hip/hip_runtime.h>
#include <math.h>

// NADE reference collapses exactly (x is binary {0,1}):
//   base[d]  = sum_h sigmoid(c[h]) * (V[d,h]+b[d])
//   delta[d] = sum_h (sigmoid(c[h]+W[h,d-1]) - sigmoid(c[h])) * (V[d,h]+b[d])
//   out[b,d] = sigmoid(base[d] + x[b,d-1]*delta[d])   (d==0: no x term)

typedef __attribute__((ext_vector_type(2))) float v2f;
typedef __attribute__((ext_vector_type(8))) float v8f;

#define B_SZ 256
#define D_SZ 784
#define H_SZ 512

// Fast sigmoid: v_exp_f32 + v_add_f32 + v_rcp_f32 (~1 ulp rcp, no IEEE div seq)
__device__ __forceinline__ float sigm(float v) {
  return __builtin_amdgcn_rcpf(1.0f + __expf(-v));
}

// One wave (32 lanes) per 16-row d-tile. Row-sums of the two D x H matrices
// are computed with V_WMMA_F32_16X16X4_F32 against an all-ones B matrix:
// D = A * ones + C  ->  every output column holds the row-sum (f32 exact path).
// No divergence anywhere in the K-loop: EXEC stays all-ones for WMMA.
__global__ __launch_bounds__(32) void nade_precompute(
    const float* __restrict__ V, const float* __restrict__ bvec,
    const float* __restrict__ W, const float* __restrict__ c,
    float* __restrict__ baseOut, float* __restrict__ deltaOut) {
  __shared__ float s0lds[H_SZ];  // sigmoid(c[h]), hoisted out of the K-loop

  const int lane = threadIdx.x;           // 0..31
  for (int h = lane; h < H_SZ; h += 32) {
    s0lds[h] = sigm(c[h]);
  }
  __syncthreads();                        // single-wave WG: lowers to S_NOP

  const int m    = lane & 15;             // row within tile (M)
  const int d0   = blockIdx.x * 16;
  const int d    = d0 + m;
  const int koff = (lane >> 4) << 1;      // lanes 0-15 -> K=0,1 ; lanes 16-31 -> K=2,3
  const float bd = bvec[d];
  // d==0 has no x term: load a valid (clamped) W column, null its contribution.
  const int   wcol  = (d > 0) ? (d - 1) : 0;
  const float dmask = (d > 0) ? 1.0f : 0.0f;

  v8f accB = {0.f,0.f,0.f,0.f,0.f,0.f,0.f,0.f};
  v8f accD = {0.f,0.f,0.f,0.f,0.f,0.f,0.f,0.f};
  const v2f ones = {1.0f, 1.0f};

  for (int k0 = 0; k0 < H_SZ; k0 += 4) {
    const int h = k0 + koff;
    // V row is contiguous in h: float2 load
    v2f v2 = *(const v2f*)(V + (size_t)d * H_SZ + h);
    float vb0 = v2.x + bd;
    float vb1 = v2.y + bd;
    float c0  = c[h], c1 = c[h + 1];
    float s00 = s0lds[h], s01 = s0lds[h + 1];
    // Column of W (stride D); unconditional, clamped for d==0
    float w0 = W[(size_t)h       * D_SZ + wcol];
    float w1 = W[(size_t)(h + 1) * D_SZ + wcol];
    float s10 = sigm(c0 + w0), s11 = sigm(c1 + w1);

    v2f a1 = { s00 * vb0, s01 * vb1 };                    // base integrand
    v2f a2 = { (s10 - s00) * vb0 * dmask,                 // delta integrand
               (s11 - s01) * vb1 * dmask };

    // 8 args: (neg_a, A, neg_b, B, c_mod, C, reuse_a, reuse_b)
    accB = __builtin_amdgcn_wmma_f32_16x16x4_f32(
        false, a1, false, ones, (short)0, accB, false, false);
    accD = __builtin_amdgcn_wmma_f32_16x16x4_f32(
        false, a2, false, ones, (short)0, accD, false, false);
  }

  // C/D layout: VGPR i holds M=i (lanes 0-15) / M=i+8 (lanes 16-31); all N
  // columns are identical (all-ones B), so lane 0 and lane 16 cover all rows.
  if (lane == 0 || lane == 16) {
    const int mb = (lane == 0) ? 0 : 8;
#pragma unroll
    for (int i = 0; i < 8; ++i) {
      baseOut[d0 + mb + i]  = accB[i];
      deltaOut[d0 + mb + i] = accD[i];
    }
  }
}

__global__ void nade_output(const float* __restrict__ x,
                            const float* __restrict__ basev,
                            const float* __restrict__ deltav,
                            float* __restrict__ out, int n) {
  int idx = blockIdx.x * blockDim.x + threadIdx.x;
  if (idx >= n) return;
  int d  = idx % D_SZ;
  int bb = idx / D_SZ;
  float xprev = (d == 0) ? 0.0f : x[(size_t)bb * D_SZ + d - 1];
  float logit = basev[d] + xprev * deltav[d];
  out[idx] = sigm(logit);
}

extern "C" void kernel_launch(void* const* d_in, const int* in_sizes, int n_in,
                              void* d_out, int out_size, void* d_ws, size_t ws_size,
                              hipStream_t stream) {
  // setup_inputs order: x (B*D), V (D*H), b (D), W (H*D), c (H) — all float32
  const float* x = (const float*)d_in[0];
  const float* V = (const float*)d_in[1];
  const float* b = (const float*)d_in[2];
  const float* W = (const float*)d_in[3];
  const float* c = (const float*)d_in[4];
  float* out = (float*)d_out;

  float* basev  = (float*)d_ws;        // D floats
  float* deltav = basev + D_SZ;        // D floats

  nade_precompute<<<D_SZ / 16, 32, 0, stream>>>(V, b, W, c, basev, deltav);

  const int n = B_SZ * D_SZ;
  nade_output<<<(n + 255) / 256, 256, 0, stream>>>(x, basev, deltav, out, n);
}